// KalmanVAE_61495341744813
// MI455X (gfx1250) — compile-verified
//
#include <hip/hip_runtime.h>
#include <math.h>

#define A_DIM 32
#define ZD    64
#define KK    8
#define HID   50
#define BB    64
#define TT    128
#define Q_VAL 0.08f
#define R_VAL 0.03f
#define SIG0  20.0f

typedef __attribute__((ext_vector_type(2))) float v2f;
typedef __attribute__((ext_vector_type(8))) float v8f;

__device__ __forceinline__ float sigmoidf(float x) { return 1.0f / (1.0f + __expf(-x)); }

// ---------------------------------------------------------------------------
// Block-cooperative GEMM via V_WMMA_F32_16X16X4_F32.
//   D(MxN) = A(MxK) * op(B)   [+ D if ACC]
//   op(B) = B (KxN, row-major, ldb) if !TB ; B^T with B stored NxK (ldb) if TB
// A/B/D may be LDS or global pointers. Caller must __syncthreads() around it.
// Wave/tile loop is forced wave-uniform (readfirstlane) so EXEC stays all-1s
// around every WMMA, per ISA 7.12 restrictions.
// ---------------------------------------------------------------------------
template <bool TB, bool ACC>
__device__ __forceinline__ void wmma_gemm(const float* __restrict__ A, int lda,
                                          const float* __restrict__ Bm, int ldb,
                                          float* __restrict__ D, int ldd,
                                          int M, int N, int Kdim)
{
    const int lane = threadIdx.x & 31;
    const int wave = __builtin_amdgcn_readfirstlane(threadIdx.x >> 5);
    const int nwav = blockDim.x >> 5;
    const int mrow = lane & 15;     // M (for A) or N (for B/D) row within tile
    const int hi   = lane >> 4;     // lane half select
    const int kh   = hi << 1;       // K offset: 0 or 2
    const int nt   = N >> 4;
    const int ntiles = (M >> 4) * nt;

    for (int tile = wave; tile < ntiles; tile += nwav) {
        const int mo = (tile / nt) << 4;
        const int no = (tile - (tile / nt) * nt) << 4;
        v8f acc;
        if (ACC) {
#pragma unroll
            for (int r = 0; r < 8; ++r)
                acc[r] = D[(size_t)(mo + (hi << 3) + r) * ldd + no + mrow];
        } else {
            acc = (v8f)0.0f;
        }
        for (int kk = 0; kk < Kdim; kk += 4) {
            v2f av, bv;
            av.x = A[(size_t)(mo + mrow) * lda + kk + kh];
            av.y = A[(size_t)(mo + mrow) * lda + kk + kh + 1];
            if (TB) {
                bv.x = Bm[(size_t)(no + mrow) * ldb + kk + kh];
                bv.y = Bm[(size_t)(no + mrow) * ldb + kk + kh + 1];
            } else {
                bv.x = Bm[(size_t)(kk + kh) * ldb + no + mrow];
                bv.y = Bm[(size_t)(kk + kh + 1) * ldb + no + mrow];
            }
            acc = __builtin_amdgcn_wmma_f32_16x16x4_f32(
                false, av, false, bv, (short)0, acc, false, false);
        }
#pragma unroll
        for (int r = 0; r < 8; ++r)
            D[(size_t)(mo + (hi << 3) + r) * ldd + no + mrow] = acc[r];
    }
}

// ---------------------------------------------------------------------------
// Block-cooperative Gauss-Jordan inverse (no pivoting; SPD+diag inputs).
// Aug: n x 2n augmented [Mtx | I]; on exit right half holds inverse.
// ---------------------------------------------------------------------------
__device__ __forceinline__ void gauss_jordan(float* Aug, float* colv, int n)
{
    const int ld = 2 * n;
    const int tid = threadIdx.x;
    for (int p = 0; p < n; ++p) {
        __syncthreads();
        const float inv_piv = 1.0f / Aug[p * ld + p];
        __syncthreads();
        for (int c = tid; c < ld; c += blockDim.x) Aug[p * ld + c] *= inv_piv;
        __syncthreads();
        for (int r = tid; r < n; r += blockDim.x) colv[r] = Aug[r * ld + p];
        __syncthreads();
        for (int i = tid; i < n * ld; i += blockDim.x) {
            const int r = i / ld;
            if (r != p) Aug[i] -= colv[r] * Aug[p * ld + (i - r * ld)];
        }
    }
    __syncthreads();
}

// ===========================================================================
// Kernel 1: LSTM, one block per batch element, sequential over T.
// hs layout: [t][b][HID]
// ===========================================================================
__global__ __launch_bounds__(256) void lstm_kernel(
    const float* __restrict__ a, const float* __restrict__ a1,
    const float* __restrict__ W_ih, const float* __restrict__ W_hh,
    const float* __restrict__ b_ih, const float* __restrict__ b_hh,
    float* __restrict__ hs)
{
    __shared__ float xt[A_DIM];
    __shared__ float hbuf[HID], cbuf[HID];
    __shared__ float gates[4 * HID];

    const int b = blockIdx.x;
    const int tid = threadIdx.x;
    if (tid < HID) { hbuf[tid] = 0.0f; cbuf[tid] = 0.0f; }
    __syncthreads();

    for (int t = 0; t < TT; ++t) {
        if (tid < A_DIM) {
            xt[tid] = (t == 0) ? a1[tid] : a[((size_t)b * TT + (t - 1)) * A_DIM + tid];
            // prefetch next timestep's input row
            if (t + 1 < TT)
                __builtin_prefetch(&a[((size_t)b * TT + t) * A_DIM + tid], 0, 3);
        }
        __syncthreads();

        if (tid < 4 * HID) {
            float s = b_ih[tid] + b_hh[tid];
            const float* wi = W_ih + (size_t)tid * A_DIM;
            const float* wh = W_hh + (size_t)tid * HID;
#pragma unroll
            for (int j = 0; j < A_DIM; ++j) s += wi[j] * xt[j];
            for (int j = 0; j < HID; ++j)  s += wh[j] * hbuf[j];
            gates[tid] = s;
        }
        __syncthreads();

        if (tid < HID) {
            const float ig = sigmoidf(gates[tid]);
            const float fg = sigmoidf(gates[HID + tid]);
            const float gg = tanhf(gates[2 * HID + tid]);
            const float og = sigmoidf(gates[3 * HID + tid]);
            const float c = fg * cbuf[tid] + ig * gg;
            const float h = og * tanhf(c);
            cbuf[tid] = c; hbuf[tid] = h;
            hs[((size_t)t * BB + b) * HID + tid] = h;
        }
        __syncthreads();
    }
}

// ===========================================================================
// Kernel 2: mixture weights (logits + softmax). One thread per (t,b).
// wbuf layout [t][b][K]; out_w layout [b][t][K].
// ===========================================================================
__global__ void mix_kernel(const float* __restrict__ hs,
                           const float* __restrict__ W_alpha,
                           const float* __restrict__ b_alpha,
                           float* __restrict__ wbuf, float* __restrict__ out_w)
{
    const int tid = blockIdx.x * blockDim.x + threadIdx.x;
    if (tid >= TT * BB) return;
    const int t = tid / BB, b = tid - t * BB;
    const float* h = hs + (size_t)tid * HID;

    float lg[KK];
    float mx = -1e30f;
#pragma unroll
    for (int k = 0; k < KK; ++k) {
        float s = b_alpha[k];
        const float* wa = W_alpha + (size_t)k * HID;
        for (int j = 0; j < HID; ++j) s += h[j] * wa[j];
        lg[k] = s; mx = fmaxf(mx, s);
    }
    float den = 0.0f;
#pragma unroll
    for (int k = 0; k < KK; ++k) { lg[k] = __expf(lg[k] - mx); den += lg[k]; }
    const float rden = 1.0f / den;
#pragma unroll
    for (int k = 0; k < KK; ++k) {
        const float v = lg[k] * rden;
        wbuf[(size_t)tid * KK + k] = v;
        out_w[((size_t)b * TT + t) * KK + k] = v;
    }
}

// ===========================================================================
// Kernel 3: Kalman filter. One block (8 waves) per batch element, seq over T.
// Stores mu_filt/sig_filt into output slots; mu_pred/sig_pred to workspace.
// ===========================================================================
__global__ __launch_bounds__(256) void filter_kernel(
    const float* __restrict__ a, const float* __restrict__ A_mat,
    const float* __restrict__ C_mat, const float* __restrict__ wbuf,
    float* __restrict__ mu_pred_ws, float* __restrict__ sig_pred_ws,
    float* __restrict__ out_mu, float* __restrict__ out_sig)
{
    __shared__ float An[ZD * ZD];        // mixed A at t+1
    __shared__ float Cs[A_DIM * ZD];     // mixed C at t
    __shared__ float SP[ZD * ZD];        // sigma_pred carry
    __shared__ float SN[ZD * ZD];        // sigma_new
    __shared__ float T3[ZD * ZD];        // temp 64x64
    __shared__ float T1[A_DIM * ZD];     // C @ SP   (32x64)
    __shared__ float T2[ZD * A_DIM];     // SP @ C^T (64x32)
    __shared__ float Kg[ZD * A_DIM];     // Kalman gain (64x32)
    __shared__ float Saug[A_DIM * 2 * A_DIM]; // [S | I] 32x64
    __shared__ float colv[ZD];
    __shared__ float mu[ZD], mupr[ZD], Amu[ZD], resid[A_DIM], munew[ZD];

    const int b = blockIdx.x;
    const int tid = threadIdx.x;

    // carry init: mu=0, mu_pred=0, sig_pred = SIG0*I
    for (int i = tid; i < ZD * ZD; i += blockDim.x)
        SP[i] = ((i >> 6) == (i & 63)) ? SIG0 : 0.0f;
    if (tid < ZD) { mu[tid] = 0.0f; mupr[tid] = 0.0f; }
    __syncthreads();

    for (int t = 0; t < TT; ++t) {
        const int tn = (t + 1 < TT) ? t + 1 : t;
        const float* wt = wbuf + ((size_t)t * BB + b) * KK;
        const float* wn = wbuf + ((size_t)tn * BB + b) * KK;
        float w0[KK], w1[KK];
#pragma unroll
        for (int k = 0; k < KK; ++k) { w0[k] = wt[k]; w1[k] = wn[k]; }
        if (tid == 0 && t + 2 < TT)   // prefetch next step's weights
            __builtin_prefetch(wbuf + ((size_t)(t + 2) * BB + b) * KK, 0, 3);

        // An = sum_k w1[k]*A_mat[k];  Cs = sum_k w0[k]*C_mat[k]
        for (int i = tid; i < ZD * ZD; i += blockDim.x) {
            float sn = 0.0f;
#pragma unroll
            for (int k = 0; k < KK; ++k) sn += w1[k] * A_mat[(size_t)k * ZD * ZD + i];
            An[i] = sn;
        }
        for (int i = tid; i < A_DIM * ZD; i += blockDim.x) {
            float sc = 0.0f;
#pragma unroll
            for (int k = 0; k < KK; ++k) sc += w0[k] * C_mat[(size_t)k * A_DIM * ZD + i];
            Cs[i] = sc;
        }
        __syncthreads();

        // T1 = Cs @ SP   (32x64, K=64)
        wmma_gemm<false, false>(Cs, ZD, SP, ZD, T1, ZD, A_DIM, ZD, ZD);
        __syncthreads();
        // Saug[:, :32] = T1 @ Cs^T  (32x32, K=64)
        wmma_gemm<true, false>(T1, ZD, Cs, ZD, Saug, 2 * A_DIM, A_DIM, A_DIM, ZD);
        __syncthreads();
        // S += R; right half = I
        for (int i = tid; i < A_DIM * 2 * A_DIM; i += blockDim.x) {
            const int r = i >> 6, c = i & 63;
            if (c < A_DIM) { if (r == c) Saug[i] += R_VAL; }
            else Saug[i] = (c - A_DIM == r) ? 1.0f : 0.0f;
        }
        gauss_jordan(Saug, colv, A_DIM);     // right half = S^-1

        // T2 = SP @ Cs^T (64x32, K=64)
        wmma_gemm<true, false>(SP, ZD, Cs, ZD, T2, A_DIM, ZD, A_DIM, ZD);
        __syncthreads();
        // Kg = T2 @ Sinv (64x32, K=32)
        wmma_gemm<false, false>(T2, A_DIM, Saug + A_DIM, 2 * A_DIM, Kg, A_DIM, ZD, A_DIM, A_DIM);
        __syncthreads();

        // Amu = (sum_k w0 A_mat[k]) @ mu   (on the fly from L2-resident A_mat)
        if (tid < ZD) {
            float s = 0.0f;
            for (int j = 0; j < ZD; ++j) {
                float aij = 0.0f;
#pragma unroll
                for (int k = 0; k < KK; ++k) aij += w0[k] * A_mat[(size_t)k * ZD * ZD + tid * ZD + j];
                s += aij * mu[j];
            }
            Amu[tid] = s;
        }
        __syncthreads();
        if (tid < A_DIM) {
            float s = 0.0f;
            for (int j = 0; j < ZD; ++j) s += Cs[tid * ZD + j] * Amu[j];
            // y[t,b] = a.reshape(T,B,A_DIM)[t,b]
            resid[tid] = a[((size_t)t * BB + b) * A_DIM + tid] - s;
        }
        __syncthreads();
        if (tid < ZD) {
            float s = 0.0f;
            for (int j = 0; j < A_DIM; ++j) s += Kg[tid * A_DIM + j] * resid[j];
            munew[tid] = Amu[tid] + s;
        }
        __syncthreads();

        // T3 = Kg @ T1 (= Kg C SP), 64x64, K=32
        wmma_gemm<false, false>(Kg, A_DIM, T1, ZD, T3, ZD, ZD, ZD, A_DIM);
        __syncthreads();

        // SN = SP - T3 ; record sig_filt -> out, sig_pred carry -> ws
        for (int i = tid; i < ZD * ZD; i += blockDim.x) {
            const float v = SP[i] - T3[i];
            SN[i] = v;
            out_sig[((size_t)b * TT + t) * (ZD * ZD) + i] = v;
            sig_pred_ws[((size_t)t * BB + b) * (ZD * ZD) + i] = SP[i];
        }
        if (tid < ZD) {
            out_mu[((size_t)b * TT + t) * ZD + tid] = munew[tid];
            mu_pred_ws[((size_t)t * BB + b) * ZD + tid] = mupr[tid];
        }
        __syncthreads();

        // carry: SP' = An @ SN @ An^T + Q ; mupr' = An @ munew ; mu' = munew
        wmma_gemm<false, false>(An, ZD, SN, ZD, T3, ZD, ZD, ZD, ZD);
        __syncthreads();
        wmma_gemm<true, false>(T3, ZD, An, ZD, SP, ZD, ZD, ZD, ZD);
        __syncthreads();
        for (int i = tid; i < ZD * ZD; i += blockDim.x)
            if ((i >> 6) == (i & 63)) SP[i] += Q_VAL;
        if (tid < ZD) {
            float s = 0.0f;
            for (int j = 0; j < ZD; ++j) s += An[tid * ZD + j] * munew[j];
            mupr[tid] = s;
            mu[tid] = munew[tid];
        }
        __syncthreads();
    }
}

// ===========================================================================
// Kernel 4: RTS smoother, in-place backwards over the output slots.
// ===========================================================================
__global__ __launch_bounds__(256) void smooth_kernel(
    const float* __restrict__ A_mat, const float* __restrict__ wbuf,
    const float* __restrict__ mu_pred_ws, const float* __restrict__ sig_pred_ws,
    float* __restrict__ out_mu, float* __restrict__ out_sig)
{
    __shared__ float An[ZD * ZD];
    __shared__ float Dm[ZD * ZD];      // sig_n - sp
    __shared__ float Jm[ZD * ZD];
    __shared__ float T1[ZD * ZD];
    __shared__ float sigN[ZD * ZD];    // carry: smoothed sigma at t+1
    __shared__ float Aug[ZD * 2 * ZD]; // [sp | I] -> [.. | sp^-1]
    __shared__ float colv[ZD];
    __shared__ float muN[ZD], dmu[ZD], mus[ZD];

    const int b = blockIdx.x;
    const int tid = threadIdx.x;

    // carry init from last filtered step (== last smoothed step)
    for (int i = tid; i < ZD * ZD; i += blockDim.x)
        sigN[i] = out_sig[((size_t)b * TT + (TT - 1)) * (ZD * ZD) + i];
    if (tid < ZD) muN[tid] = out_mu[((size_t)b * TT + (TT - 1)) * ZD + tid];
    __syncthreads();

    for (int t = TT - 2; t >= 0; --t) {
        const float* wn = wbuf + ((size_t)(t + 1) * BB + b) * KK;
        float w1[KK];
#pragma unroll
        for (int k = 0; k < KK; ++k) w1[k] = wn[k];

        const float* sp = sig_pred_ws + ((size_t)(t + 1) * BB + b) * (ZD * ZD);
        const float* sf = out_sig + ((size_t)b * TT + t) * (ZD * ZD);

        // prefetch next (earlier) iteration's streamed rows while we compute
        if (t > 0) {
            const float* spn = sig_pred_ws + ((size_t)t * BB + b) * (ZD * ZD);
            const float* sfn = out_sig + ((size_t)b * TT + (t - 1)) * (ZD * ZD);
            for (int i = tid; i < ZD; i += blockDim.x) {
                __builtin_prefetch(spn + (size_t)i * ZD, 0, 3);
                __builtin_prefetch(sfn + (size_t)i * ZD, 0, 3);
            }
        }

        // An = sum_k w1 A_mat[k]; Dm = sigN - sp ; Aug = [sp | I]
        for (int i = tid; i < ZD * ZD; i += blockDim.x) {
            float sa = 0.0f;
#pragma unroll
            for (int k = 0; k < KK; ++k) sa += w1[k] * A_mat[(size_t)k * ZD * ZD + i];
            An[i] = sa;
            const float spv = sp[i];
            Dm[i] = sigN[i] - spv;
            const int r = i >> 6, c = i & 63;
            Aug[r * (2 * ZD) + c] = spv;
            Aug[r * (2 * ZD) + ZD + c] = (r == c) ? 1.0f : 0.0f;
        }
        gauss_jordan(Aug, colv, ZD);         // right half = sp^-1

        // T1 = sf @ An^T          (sf read from global; L2 resident)
        wmma_gemm<true, false>(sf, ZD, An, ZD, T1, ZD, ZD, ZD, ZD);
        __syncthreads();
        // J = T1 @ sp^-1
        wmma_gemm<false, false>(T1, ZD, Aug + ZD, 2 * ZD, Jm, ZD, ZD, ZD, ZD);
        __syncthreads();
        // T1 = J @ Dm
        wmma_gemm<false, false>(Jm, ZD, Dm, ZD, T1, ZD, ZD, ZD, ZD);
        __syncthreads();
        // Dm = T1 @ J^T   (reuse Dm as output; its input role is done)
        wmma_gemm<true, false>(T1, ZD, Jm, ZD, Dm, ZD, ZD, ZD, ZD);
        __syncthreads();

        // mu path
        if (tid < ZD)
            dmu[tid] = muN[tid] - mu_pred_ws[((size_t)(t + 1) * BB + b) * ZD + tid];
        __syncthreads();
        if (tid < ZD) {
            float s = 0.0f;
            for (int j = 0; j < ZD; ++j) s += Jm[tid * ZD + j] * dmu[j];
            mus[tid] = out_mu[((size_t)b * TT + t) * ZD + tid] + s;
        }
        __syncthreads();

        // sig_s = sf + Dm ; commit (in place) and update carries
        for (int i = tid; i < ZD * ZD; i += blockDim.x) {
            const float v = sf[i] + Dm[i];
            sigN[i] = v;
            out_sig[((size_t)b * TT + t) * (ZD * ZD) + i] = v;
        }
        if (tid < ZD) {
            muN[tid] = mus[tid];
            out_mu[((size_t)b * TT + t) * ZD + tid] = mus[tid];
        }
        __syncthreads();
    }
}

// ===========================================================================
extern "C" void kernel_launch(void* const* d_in, const int* in_sizes, int n_in,
                              void* d_out, int out_size, void* d_ws, size_t ws_size,
                              hipStream_t stream)
{
    const float* a       = (const float*)d_in[0];
    const float* a1      = (const float*)d_in[1];
    const float* W_ih    = (const float*)d_in[2];
    const float* W_hh    = (const float*)d_in[3];
    const float* b_ih    = (const float*)d_in[4];
    const float* b_hh    = (const float*)d_in[5];
    const float* W_alpha = (const float*)d_in[6];
    const float* b_alpha = (const float*)d_in[7];
    const float* A_mat   = (const float*)d_in[8];
    const float* C_mat   = (const float*)d_in[9];

    float* ws = (float*)d_ws;
    float* hs       = ws;                                   // T*B*HID     = 409600
    float* wbuf     = hs + (size_t)TT * BB * HID;           // T*B*K       = 65536
    float* mu_pred  = wbuf + (size_t)TT * BB * KK;          // T*B*Z       = 524288
    float* sig_pred = mu_pred + (size_t)TT * BB * ZD;       // T*B*Z*Z     = 33554432

    float* out    = (float*)d_out;
    float* out_mu = out;                                    // B*T*Z
    float* out_sig = out + (size_t)BB * TT * ZD;            // B*T*Z*Z
    float* out_w  = out_sig + (size_t)BB * TT * ZD * ZD;    // B*T*K

    lstm_kernel<<<BB, 256, 0, stream>>>(a, a1, W_ih, W_hh, b_ih, b_hh, hs);
    mix_kernel<<<(TT * BB + 255) / 256, 256, 0, stream>>>(hs, W_alpha, b_alpha, wbuf, out_w);
    filter_kernel<<<BB, 256, 0, stream>>>(a, A_mat, C_mat, wbuf,
                                          mu_pred, sig_pred, out_mu, out_sig);
    smooth_kernel<<<BB, 256, 0, stream>>>(A_mat, wbuf, mu_pred, sig_pred,
                                          out_mu, out_sig);
}